// SegmentationDiceCoefficient_575525617708
// MI455X (gfx1250) — compile-verified
//
#include <hip/hip_runtime.h>
#include <hip/hip_bf16.h>

// Segmentation dice: B=32, C=4, H=W=512.
// d_in[0] = true_masks (int32, B*H*W), d_in[1] = out (float32, B*C*H*W)
// d_out   = dice[1:4] (3 floats)
// HBM-streaming reduction: ~168 MB read once -> ~7.2 us floor at 23.3 TB/s.

#define HW      (512 * 512)       // 262144 pixels per (b, c) plane
#define CHW     (4 * HW)          // per-batch stride
#define NPIX    (32 * HW)         // 8,388,608 total pixels
#define NGRP    (NPIX / 4)        // 2,097,152 float4 groups
#define NBLK    1024
#define NTHR    256
// NGRP / (NBLK*NTHR) = 8 iterations -> 32 pixels/thread -> 8-bit packed
// per-class counters cannot overflow (max 32 < 255).

typedef float f32x4 __attribute__((ext_vector_type(4)));
typedef int   i32x4 __attribute__((ext_vector_type(4)));

__global__ void dice_zero_ws(unsigned int* __restrict__ ws) {
    if (threadIdx.x < 12) ws[threadIdx.x] = 0u;
}

__global__ __launch_bounds__(NTHR)
void dice_count(const float* __restrict__ logits,
                const int* __restrict__ masks,
                unsigned int* __restrict__ ws) {
    __shared__ unsigned int hist[12];   // [0..3]=tp [4..7]=pred [8..11]=true
    if (threadIdx.x < 12) hist[threadIdx.x] = 0u;
    __syncthreads();

    // Packed per-thread counters: 8 bits per class, class c in bits [8c+7:8c].
    unsigned int tp_pack = 0u, pr_pack = 0u, tr_pack = 0u;

    const unsigned int stride = NBLK * NTHR;
    for (unsigned int g = blockIdx.x * NTHR + threadIdx.x; g < NGRP; g += stride) {
        const unsigned int p = g << 2;          // base pixel index (16B aligned)
        const unsigned int b = p >> 18;         // p / HW
        const unsigned int r = p & (HW - 1);    // p % HW
        const float* base = logits + (size_t)b * CHW + r;

        // 4 pixels x 4 classes: five coalesced 128-bit non-temporal loads
        const f32x4 c0 = __builtin_nontemporal_load((const f32x4*)(base));
        const f32x4 c1 = __builtin_nontemporal_load((const f32x4*)(base + HW));
        const f32x4 c2 = __builtin_nontemporal_load((const f32x4*)(base + 2 * HW));
        const f32x4 c3 = __builtin_nontemporal_load((const f32x4*)(base + 3 * HW));
        const i32x4 mm = __builtin_nontemporal_load((const i32x4*)(masks + p));

        // Branchless first-occurrence argmax + packed counter updates.
#define DICE_PIX(a0, a1, a2, a3, tmask)                                        \
        {                                                                      \
            const int   pr01 = ((a1) > (a0)) ? 1 : 0;                          \
            const float mx01 = fmaxf((a0), (a1));                              \
            const int   pr23 = ((a3) > (a2)) ? 3 : 2;                          \
            const float mx23 = fmaxf((a2), (a3));                              \
            const int   pr   = (mx23 > mx01) ? pr23 : pr01;                    \
            const unsigned int prs = (unsigned int)pr << 3;                    \
            pr_pack += 1u << prs;                                              \
            tr_pack += 1u << ((unsigned int)(tmask) << 3);                     \
            tp_pack += (unsigned int)(pr == (tmask)) << prs;                   \
        }

        DICE_PIX(c0.x, c1.x, c2.x, c3.x, mm.x);
        DICE_PIX(c0.y, c1.y, c2.y, c3.y, mm.y);
        DICE_PIX(c0.z, c1.z, c2.z, c3.z, mm.z);
        DICE_PIX(c0.w, c1.w, c2.w, c3.w, mm.w);
#undef DICE_PIX
    }

    // Unpack and reduce: 12 LDS atomics/thread, then 12 global atomics/block.
#pragma unroll
    for (int c = 0; c < 4; ++c) {
        const unsigned int sh = (unsigned int)c << 3;
        const unsigned int tpv = (tp_pack >> sh) & 0xFFu;
        const unsigned int prv = (pr_pack >> sh) & 0xFFu;
        const unsigned int trv = (tr_pack >> sh) & 0xFFu;
        if (tpv) atomicAdd(&hist[c], tpv);
        if (prv) atomicAdd(&hist[4 + c], prv);
        if (trv) atomicAdd(&hist[8 + c], trv);
    }
    __syncthreads();
    if (threadIdx.x < 12) {
        const unsigned int v = hist[threadIdx.x];
        if (v) atomicAdd(&ws[threadIdx.x], v);
    }
}

__global__ void dice_finalize(const unsigned int* __restrict__ ws,
                              float* __restrict__ out) {
    const int c = (int)threadIdx.x + 1;       // classes 1..3 (skip background)
    if (c < 4) {
        const float tp = (float)ws[c];
        const float pc = (float)ws[4 + c];
        const float tc = (float)ws[8 + c];
        out[c - 1] = 2.0f * tp / (pc + tc);
    }
}

extern "C" void kernel_launch(void* const* d_in, const int* in_sizes, int n_in,
                              void* d_out, int out_size, void* d_ws, size_t ws_size,
                              hipStream_t stream) {
    const int*   masks  = (const int*)d_in[0];    // true_masks, B*H*W int32
    const float* logits = (const float*)d_in[1];  // out, B*C*H*W float32
    float*        dice  = (float*)d_out;          // 3 floats
    unsigned int* ws    = (unsigned int*)d_ws;    // 12 counters

    dice_zero_ws<<<1, 32, 0, stream>>>(ws);
    dice_count<<<NBLK, NTHR, 0, stream>>>(logits, masks, ws);
    dice_finalize<<<1, 32, 0, stream>>>(ws, dice);
    (void)in_sizes; (void)n_in; (void)out_size; (void)ws_size;
}